// ParabolicPool2D_V2_TL_40106404610422
// MI455X (gfx1250) — compile-verified
//
#include <hip/hip_runtime.h>
#include <hip/hip_bf16.h>
#include <stdint.h>

// Problem constants (from reference): f[32,96,128,128] f32, t[96] f32, ks=3, stride=2
#define B_   32
#define C_   96
#define H_   128
#define W_   128
#define HOUT 63
#define WOUT 63
#define ROWS_PER_BAND 9
#define NBANDS 7                        // 7 * 9 = 63 output rows
#define TILE_ROWS (2 * ROWS_PER_BAND + 1)   // 19 input rows per band
#define TILE_ELEMS (TILE_ROWS * W_)         // 2432 floats = 9728 B of LDS

#if defined(__has_builtin)
#  if __has_builtin(__builtin_amdgcn_tensor_load_to_lds) && \
      __has_builtin(__builtin_amdgcn_s_wait_tensorcnt)
#    define USE_TDM 1
#  endif
#endif

typedef __attribute__((ext_vector_type(4))) unsigned int v4u;
typedef __attribute__((ext_vector_type(8))) int          v8i;
typedef __attribute__((ext_vector_type(4))) int          v4i;

__global__ __launch_bounds__(256)
void ParabolicPool2D_kernel(const float* __restrict__ f,
                            const float* __restrict__ t,
                            float* __restrict__ out)
{
    __shared__ __align__(16) float smem[TILE_ELEMS];

    const int band = blockIdx.x % NBANDS;
    const int bc   = blockIdx.x / NBANDS;          // b*C + c
    const int c    = bc % C_;
    const int r0   = band * ROWS_PER_BAND;         // first output row of band
    const int in_row0 = r0 * 2;                    // first input row needed

    // Byte address of tile start (tile spans the full 128-wide row, contiguous rows)
    const uint64_t gaddr = (uint64_t)(uintptr_t)f +
        (((uint64_t)bc * H_ + (uint64_t)in_row0) * W_) * sizeof(float);

#ifdef USE_TDM
    if (threadIdx.x < 32) {                        // one wave issues the TDM DMA
        const uint32_t lds = (uint32_t)(uintptr_t)smem;   // flat low 32 bits == LDS offset

        // ---- D# group 0 (128b): count=1 | lds_addr | global_addr[56:0] | type=2 ----
        v4u g0;
        g0.x = 1u;                                           // count=1, user descriptor
        g0.y = lds;                                          // lds_addr
        g0.z = (uint32_t)(gaddr & 0xffffffffull);            // global_addr[31:0]
        g0.w = (uint32_t)((gaddr >> 32) & 0x1ffffffull)      // global_addr[56:32]
             | (2u << 30);                                   // type = 2 ("image")

        // ---- D# group 1 (256b): 2D tensor W_ x (B*C*H), tile W_ x TILE_ROWS ----
        // bits: [17:16]=data_size(2 => 4B) [79:48]=tensor_dim0 [111:80]=tensor_dim1
        //       [127:112]=tile_dim0 [143:128]=tile_dim1 [159:144]=tile_dim2(0)
        //       [207:160]=tensor_dim0_stride [255:208]=tensor_dim1_stride(0)
        const uint32_t tdim1 = B_ * C_ * H_;                 // 393216 rows total
        v8i g1;
        g1[0] = (int)(2u << 16);                             // data_size = 4 bytes
        g1[1] = (int)((uint32_t)(W_ & 0xffff) << 16);        // tensor_dim0[15:0]
        g1[2] = (int)((uint32_t)(W_ >> 16)                   // tensor_dim0[31:16]
                    | ((tdim1 & 0xffffu) << 16));            // tensor_dim1[15:0]
        g1[3] = (int)((tdim1 >> 16)                          // tensor_dim1[31:16]
                    | ((uint32_t)W_ << 16));                 // tile_dim0 = 128
        g1[4] = (int)(uint32_t)TILE_ROWS;                    // tile_dim1 = 19, tile_dim2 = 0
        g1[5] = (int)(uint32_t)W_;                           // tensor_dim0_stride = 128 elems
        g1[6] = 0;                                           // stride hi / dim1_stride lo
        g1[7] = 0;

        v4i gz4 = {0, 0, 0, 0};                              // groups 2/3 unused (2D)
        v8i gz8 = {0, 0, 0, 0, 0, 0, 0, 0};                  // extra operand (6-arg form)
        __builtin_amdgcn_tensor_load_to_lds(g0, g1, gz4, gz4, gz8, 0);
        __builtin_amdgcn_s_wait_tensorcnt(0);                // DMA complete -> LDS valid
    }
    __syncthreads();
    asm volatile("" ::: "memory");   // keep the optimizer from assuming smem is unwritten
#else
    {
        const float* src = (const float*)(uintptr_t)gaddr;   // 19*128 contiguous floats
        for (int i = threadIdx.x; i < TILE_ELEMS; i += blockDim.x)
            smem[i] = src[i];
    }
    __syncthreads();
#endif

    // Parabolic SE has 3 distinct weights: center 0, edge -1/(4t), corner -2/(4t)
    const float w1 = -1.0f / (4.0f * t[c]);
    const float w2 = w1 + w1;

    float* __restrict__ oplane = out + ((size_t)bc * HOUT + r0) * WOUT;

    for (int idx = threadIdx.x; idx < ROWS_PER_BAND * WOUT; idx += blockDim.x) {
        const int r = idx / WOUT;
        const int j = idx - r * WOUT;
        const float* p0 = &smem[(2 * r) * W_ + 2 * j];
        const float* p1 = p0 + W_;
        const float* p2 = p1 + W_;
        float a00 = p0[0], a01 = p0[1], a02 = p0[2];
        float a10 = p1[0], a11 = p1[1], a12 = p1[2];
        float a20 = p2[0], a21 = p2[1], a22 = p2[2];
        float edge = fmaxf(fmaxf(a01, a10), fmaxf(a12, a21));
        float corn = fmaxf(fmaxf(a00, a02), fmaxf(a20, a22));
        oplane[(size_t)r * WOUT + j] = fmaxf(a11, fmaxf(edge + w1, corn + w2));
    }
}

extern "C" void kernel_launch(void* const* d_in, const int* in_sizes, int n_in,
                              void* d_out, int out_size, void* d_ws, size_t ws_size,
                              hipStream_t stream) {
    (void)in_sizes; (void)n_in; (void)d_ws; (void)ws_size; (void)out_size;
    const float* f = (const float*)d_in[0];
    const float* t = (const float*)d_in[1];
    float* out = (float*)d_out;

    dim3 grid(B_ * C_ * NBANDS);   // 32*96*7 = 21504 blocks
    dim3 block(256);               // 8 wave32s
    ParabolicPool2D_kernel<<<grid, block, 0, stream>>>(f, t, out);
}